// Route_33509334843729
// MI455X (gfx1250) — compile-verified
//
#include <hip/hip_runtime.h>
#include <hip/hip_bf16.h>
#include <math.h>

// ---------------- problem dims ----------------
#define B_    180   // batch
#define NIN_  8     // in-capsule dim (GEMM K)
#define S_    1152  // number of input capsules
#define O_    10    // out caps num
#define K_    16    // out caps size
#define OK_   160   // O_*K_  (GEMM M)
#define MT_   10    // M tiles of 16
#define NT_   12    // N tiles of 16 over B (last partial: 180 = 11*16+4)
#define SCH_  8     // s-chunks for the s_j reduction
#define SCHLEN_ (S_ / SCH_)   // 144

typedef float v2f __attribute__((ext_vector_type(2)));
typedef float v4f __attribute__((ext_vector_type(4)));
typedef float v8f __attribute__((ext_vector_type(8)));

// ---------------- workspace layout (in floats) ----------------
// uhat  : [S][B][OK]  = 33,177,600 floats (132.7 MB, fits the 192MB L2)
// bij   : [S][16]     (O padded to 16)
// c     : [S][16]
// v     : [B][OK]
// psum  : [SCH][B][OK] partial s_j sums
#define UHAT_OFF  0ull
#define UHAT_SZ   ((size_t)S_ * B_ * OK_)
#define BIJ_OFF   (UHAT_OFF + UHAT_SZ)
#define BIJ_SZ    ((size_t)S_ * 16)
#define C_OFF     (BIJ_OFF + BIJ_SZ)
#define C_SZ      ((size_t)S_ * 16)
#define V_OFF     (C_OFF + C_SZ)
#define V_SZ      ((size_t)B_ * OK_)
#define PSUM_OFF  (V_OFF + V_SZ)
#define PSUM_SZ   ((size_t)SCH_ * B_ * OK_)

// =====================================================================
// Kernel 1: u_hat[s][b][ok] = sum_n W[s,ok,n] * x[b,n,s]
// One block (12 waves) per s. W_s (160x8 f32 = 5120B) is staged into LDS
// once via GLOBAL_LOAD_ASYNC_TO_LDS_B128 (ASYNCcnt), overlapped with the
// per-wave x loads; then each wave runs 10 M-tiles x 2 K-steps of
// V_WMMA_F32_16X16X4_F32 with A-operands from ds_load_b64.
// =====================================================================
__global__ __launch_bounds__(384) void uhat_wmma_kernel(
    const float* __restrict__ x,      // [B][NIN][S]
    const float* __restrict__ W,      // [S][OK][NIN]
    float* __restrict__ uhat)         // [S][B][OK]
{
    __shared__ __align__(16) float sW[OK_ * NIN_];   // 5120 bytes

    const int s     = blockIdx.x;
    const int t     = threadIdx.x;
    const int lane  = t & 31;
    const int ntile = t >> 5;                        // wave id = N tile (0..11)

    // ---- async stage of W_s into LDS: 320 lanes x 16B ----
    if (t < (OK_ * NIN_) / 4) {
        const float* gsrc = W + (size_t)s * (OK_ * NIN_) + (size_t)t * 4;
        unsigned ldsoff = (unsigned)(size_t)(&sW[t * 4]);
        asm volatile("global_load_async_to_lds_b128 %0, %1, off"
                     :: "v"(ldsoff), "v"(gsrc)
                     : "memory");
    }

    // ---- B operand (4x16 f32), overlapped with the async staging ----
    // b[j] = x[colb, n = kstep*4 + khalf + j, s]
    const int colb  = ntile * 16 + (lane & 15);
    const int khalf = (lane < 16) ? 0 : 2;
    v2f bop[2];
#pragma unroll
    for (int ks = 0; ks < 2; ++ks) {
        const int n0 = ks * 4 + khalf;
        float b0 = 0.0f, b1 = 0.0f;
        if (colb < B_) {
            const size_t base = (size_t)colb * (NIN_ * S_) + (size_t)s;
            b0 = x[base + (size_t)n0       * S_];
            b1 = x[base + (size_t)(n0 + 1) * S_];
        }
        bop[ks][0] = b0;
        bop[ks][1] = b1;
    }

    // wait own async ops, then make staged W_s visible block-wide
    asm volatile("s_wait_asynccnt 0" ::: "memory");
    __syncthreads();

    const bool full_tile = (ntile < NT_ - 1);        // wave-uniform
    const int  okoff     = (lane < 16) ? 0 : 8;
    float* const dstb = uhat + ((size_t)s * B_ + colb) * OK_;

#pragma unroll
    for (int mt = 0; mt < MT_; ++mt) {
        const int row = mt * 16 + (lane & 15);       // A-matrix ok-row
        v8f acc = {0.f, 0.f, 0.f, 0.f, 0.f, 0.f, 0.f, 0.f};
#pragma unroll
        for (int ks = 0; ks < 2; ++ks) {
            // A operand from LDS: contiguous, 8B aligned -> ds_load_b64
            const v2f a = *(const v2f*)(&sW[row * NIN_ + ks * 4 + khalf]);
            acc = __builtin_amdgcn_wmma_f32_16x16x4_f32(
                false, a, false, bop[ks], (short)0, acc, false, false);
        }
        // D layout: VGPR i -> M = i + (lane<16?0:8), N = lane%16
        float* dst = dstb + mt * 16 + okoff;
        v4f lo = {acc[0], acc[1], acc[2], acc[3]};
        v4f hi = {acc[4], acc[5], acc[6], acc[7]};
        if (full_tile) {
            *(v4f*)(dst)     = lo;
            *(v4f*)(dst + 4) = hi;
        } else if (colb < B_) {                      // ragged last N-tile
            *(v4f*)(dst)     = lo;
            *(v4f*)(dst + 4) = hi;
        }
    }
}

// =====================================================================
// Kernel: zero-init b_ij (deterministic per launch)
// =====================================================================
__global__ void init_bij_kernel(float* __restrict__ bij, int n)
{
    const int i = blockIdx.x * blockDim.x + threadIdx.x;
    if (i < n) bij[i] = 0.0f;
}

// =====================================================================
// Kernel: c[s,:] = softmax(b_ij[s,:]) over O_ entries (stride-16 padded)
// =====================================================================
__global__ void softmax_kernel(const float* __restrict__ bij,
                               float* __restrict__ c)
{
    const int s = blockIdx.x * blockDim.x + threadIdx.x;
    if (s >= S_) return;
    float vals[O_];
    float m = -3.4e38f;
#pragma unroll
    for (int o = 0; o < O_; ++o) {
        vals[o] = bij[s * 16 + o];
        m = fmaxf(m, vals[o]);
    }
    float sum = 0.0f;
#pragma unroll
    for (int o = 0; o < O_; ++o) {
        vals[o] = expf(vals[o] - m);
        sum += vals[o];
    }
    const float inv = 1.0f / sum;
#pragma unroll
    for (int o = 0; o < O_; ++o) c[s * 16 + o] = vals[o] * inv;
}

// =====================================================================
// Kernel: partial s_j over one s-chunk.
// psum[ch][b][ok] = sum_{s in chunk} c[s,o] * u_hat[s,b,ok]
// Grid = B_*SCH_ blocks of 160 threads (thread = ok) -> 1440 blocks.
// =====================================================================
__global__ __launch_bounds__(160) void sj_partial_kernel(
    const float* __restrict__ uhat,   // [S][B][OK]
    const float* __restrict__ c,      // [S][16]
    float* __restrict__ psum)         // [SCH][B][OK]
{
    const int b  = blockIdx.x % B_;
    const int ch = blockIdx.x / B_;
    const int t  = threadIdx.x;       // ok index
    const int o  = t / K_;

    float acc = 0.0f;
    const int s0 = ch * SCHLEN_;
    for (int i = 0; i < SCHLEN_; ++i) {
        const int s = s0 + i;
        acc += c[s * 16 + o] * uhat[((size_t)s * B_ + b) * OK_ + t];
    }
    psum[((size_t)ch * B_ + b) * OK_ + t] = acc;
}

// =====================================================================
// Kernel: fold partials -> s_j, then squash -> v[b,ok].
// squash: msq[b,k] = sum_o s_j^2 ; v = sqrt(msq)/(1+sqrt(msq)) * s_j/msq
// =====================================================================
__global__ __launch_bounds__(160) void squash_kernel(
    const float* __restrict__ psum,   // [SCH][B][OK]
    float* __restrict__ vws,          // [B][OK]
    float* __restrict__ out,          // [B][OK] final output
    int write_out)
{
    __shared__ float sj[OK_];
    const int b = blockIdx.x;
    const int t = threadIdx.x;        // ok index
    const int k = t % K_;

    float acc = 0.0f;
#pragma unroll
    for (int ch = 0; ch < SCH_; ++ch) {
        acc += psum[((size_t)ch * B_ + b) * OK_ + t];
    }
    sj[t] = acc;
    __syncthreads();

    float msq = 0.0f;
#pragma unroll
    for (int oo = 0; oo < O_; ++oo) {
        const float e = sj[oo * K_ + k];
        msq += e * e;
    }
    const float mod = sqrtf(msq);
    const float val = (mod / (1.0f + mod)) * (acc / msq);

    vws[b * OK_ + t] = val;
    if (write_out) out[b * OK_ + t] = val;
}

// =====================================================================
// Kernel: b_ij[s,o] += (1/B) * sum_{b,k} u_hat[s,b,ok] * v[b,ok]
// One wave32 per (s,o); shuffle-reduce across 32 lanes (warpSize==32).
// =====================================================================
__global__ __launch_bounds__(256) void uv_update_kernel(
    const float* __restrict__ uhat,   // [S][B][OK]
    const float* __restrict__ vws,    // [B][OK]
    float* __restrict__ bij)          // [S][16]
{
    const int lane = threadIdx.x & 31;
    const int wid  = blockIdx.x * (blockDim.x >> 5) + (threadIdx.x >> 5);
    if (wid >= S_ * O_) return;       // wave-uniform guard
    const int s = wid / O_;
    const int o = wid % O_;
    const int k    = lane & 15;
    const int half = lane >> 4;
    const int ok   = o * K_ + k;

    float acc = 0.0f;
    for (int b = half; b < B_; b += 2) {
        acc += uhat[((size_t)s * B_ + b) * OK_ + ok] * vws[b * OK_ + ok];
    }
#pragma unroll
    for (int off = 16; off >= 1; off >>= 1) {
        acc += __shfl_xor(acc, off, 32);
    }
    if (lane == 0) bij[s * 16 + o] += acc * (1.0f / (float)B_);
}

// =====================================================================
// Host-side launch (graph-capture safe: only kernel launches on `stream`)
// =====================================================================
extern "C" void kernel_launch(void* const* d_in, const int* in_sizes, int n_in,
                              void* d_out, int out_size, void* d_ws, size_t ws_size,
                              hipStream_t stream)
{
    const float* x = (const float*)d_in[0];   // [180][8][1152]
    const float* W = (const float*)d_in[1];   // [1152][10][16][8]
    float* out = (float*)d_out;               // [180][10][16][1] -> flat [180][160]
    float* ws  = (float*)d_ws;

    float* uhat = ws + UHAT_OFF;
    float* bij  = ws + BIJ_OFF;
    float* c    = ws + C_OFF;
    float* v    = ws + V_OFF;
    float* psum = ws + PSUM_OFF;

    // 1) b_ij = 0
    init_bij_kernel<<<(S_ * 16 + 255) / 256, 256, 0, stream>>>(bij, S_ * 16);

    // 2) u_hat via f32 WMMA + async LDS staging of W: one block per s
    uhat_wmma_kernel<<<S_, 384, 0, stream>>>(x, W, uhat);

    // 3) three routing iterations
    for (int it = 0; it < 3; ++it) {
        softmax_kernel<<<(S_ + 255) / 256, 256, 0, stream>>>(bij, c);
        sj_partial_kernel<<<B_ * SCH_, 160, 0, stream>>>(uhat, c, psum);
        squash_kernel<<<B_, 160, 0, stream>>>(psum, v, out, (it == 2) ? 1 : 0);
        if (it < 2) {
            uv_update_kernel<<<(S_ * O_ + 7) / 8, 256, 0, stream>>>(uhat, v, bij);
        }
    }
}